// GraphAttnBlock_37924561223997
// MI455X (gfx1250) — compile-verified
//
#include <hip/hip_runtime.h>
#include <math.h>

typedef __attribute__((ext_vector_type(2))) float v2f;
typedef __attribute__((ext_vector_type(8))) float v8f;

#define D1 256          // H*C
#define H4 4
#define C64 64

// ---------------------------------------------------------------------------
// WMMA fp32 GEMM: C[M,256] = A[M,K] @ B[K,256] (+bias), K in {128,256}, M%16==0
// Block: 256 threads = 8 waves. Block tile = 128 rows x 16 cols.
// B strip (K x 16) staged in LDS, shared by all 8 waves.
// A fragment: 16x4 layout -> lane (m = lane&15, hi = lane>>4) holds
//   {A[m, k+2*hi], A[m, k+2*hi+1]}  -> one contiguous 8B load.
// B fragment: lane (n = lane&15, hi) holds {B[k+2*hi, n], B[k+2*hi+1, n]}.
// ---------------------------------------------------------------------------
__global__ __launch_bounds__(256) void wmma_gemm_f32(
    const float* __restrict__ A, const float* __restrict__ B,
    const float* __restrict__ bias, float* __restrict__ C,
    int Mtiles, int K)
{
    __shared__ float Bs[D1 * 16];          // up to 256x16 fp32 = 16 KB
    const int tid   = threadIdx.x;
    const int ntile = blockIdx.y;

    for (int i = tid; i < K * 16; i += 256) {
        int k = i >> 4, n = i & 15;
        Bs[i] = B[(size_t)k * D1 + ntile * 16 + n];
    }
    __syncthreads();

    const int wave = tid >> 5;
    const int lane = tid & 31;
    const int mt   = blockIdx.x * 8 + wave;     // 16-row tile id
    if (mt >= Mtiles) return;                   // whole wave exits together

    const int m16  = lane & 15;
    const int hi   = lane >> 4;                 // 0 or 1
    const int ncol = ntile * 16 + m16;
    const float* arow = A + (size_t)(mt * 16 + m16) * K;

    v8f acc = {};
    #pragma unroll 4
    for (int k = 0; k < K; k += 4) {
        v2f a = *(const v2f*)(arow + k + 2 * hi);
        v2f b;
        b.x = Bs[(k + 2 * hi) * 16 + m16];
        b.y = Bs[(k + 2 * hi + 1) * 16 + m16];
        acc = __builtin_amdgcn_wmma_f32_16x16x4_f32(
                  false, a, false, b, (short)0, acc, false, false);
    }

    const float badd = bias ? bias[ncol] : 0.0f;
    #pragma unroll
    for (int r = 0; r < 8; ++r) {
        C[(size_t)(mt * 16 + r + 8 * hi) * D1 + ncol] = acc[r] + badd;
    }
}

// ---------------------------------------------------------------------------
// a_s[n,h] = sum_c XS[n,h,c]*atts[h,c].  One wave per node; lane strip of 8
// channels, head = lane>>3, reduce within 8-lane groups.
// ---------------------------------------------------------------------------
__global__ __launch_bounds__(256) void attn_src_scores(
    const float* __restrict__ XS, const float* __restrict__ atts,
    float* __restrict__ AS, int n_nodes)
{
    int wid  = (int)((blockIdx.x * (size_t)blockDim.x + threadIdx.x) >> 5);
    int lane = threadIdx.x & 31;
    if (wid >= n_nodes) return;
    const float* xr = XS + (size_t)wid * D1 + lane * 8;
    const float* ar = atts + lane * 8;
    float s = 0.f;
    #pragma unroll
    for (int i = 0; i < 8; ++i) s += xr[i] * ar[i];
    s += __shfl_xor(s, 1, 32);
    s += __shfl_xor(s, 2, 32);
    s += __shfl_xor(s, 4, 32);
    if ((lane & 7) == 0) AS[(size_t)wid * H4 + (lane >> 3)] = s;
}

// V[k,h] = sum_c Wdst[k, h*64+c] * attd[h,c]   (collapses the Wdst GEMM)
__global__ void collapse_att(const float* __restrict__ W,
                             const float* __restrict__ att,
                             float* __restrict__ V, int K)
{
    int idx = blockIdx.x * blockDim.x + threadIdx.x;
    if (idx >= K * H4) return;
    int k = idx >> 2, h = idx & 3;
    float s = 0.f;
    #pragma unroll 8
    for (int c = 0; c < C64; ++c)
        s += W[(size_t)k * D1 + h * C64 + c] * att[h * C64 + c];
    V[idx] = s;
}

// a_d = X @ V  (N x K @ K x 4). One wave per node, full-wave reduction.
__global__ __launch_bounds__(256) void gemv_ad(
    const float* __restrict__ X, const float* __restrict__ V,
    float* __restrict__ AD, int n_nodes, int K)
{
    int wid  = (int)((blockIdx.x * (size_t)blockDim.x + threadIdx.x) >> 5);
    int lane = threadIdx.x & 31;
    if (wid >= n_nodes) return;
    int per = K >> 5;                       // 4 (K=128) or 8 (K=256)
    const float* xr = X + (size_t)wid * K + lane * per;
    float a0 = 0.f, a1 = 0.f, a2 = 0.f, a3 = 0.f;
    for (int i = 0; i < per; ++i) {
        float xv = xr[i];
        const float* v = V + (size_t)(lane * per + i) * H4;
        a0 += xv * v[0]; a1 += xv * v[1]; a2 += xv * v[2]; a3 += xv * v[3];
    }
    #pragma unroll
    for (int d = 16; d >= 1; d >>= 1) {
        a0 += __shfl_xor(a0, d, 32); a1 += __shfl_xor(a1, d, 32);
        a2 += __shfl_xor(a2, d, 32); a3 += __shfl_xor(a3, d, 32);
    }
    if (lane == 0) {
        float* o = AD + (size_t)wid * H4;
        o[0] = a0; o[1] = a1; o[2] = a2; o[3] = a3;
    }
}

__global__ void fill_f32(float* __restrict__ p, float v, long n)
{
    long i = (long)blockIdx.x * blockDim.x + threadIdx.x;
    long s = (long)gridDim.x * blockDim.x;
    for (; i < n; i += s) p[i] = v;
}

// sign-aware float atomic max built from integer atomics (lowers to
// global_atomic_max_i32 / global_atomic_min_u32 — no CAS loop)
__device__ __forceinline__ void atomicMaxF(float* addr, float val)
{
    if (val >= 0.f)
        atomicMax((int*)addr, __float_as_int(val));
    else
        atomicMin((unsigned int*)addr, __float_as_uint(val));
}

// pass 1: e = leaky_relu(a_s[src]+a_d[dst]); segment max over dst
__global__ void edge_max(const int* __restrict__ src, const int* __restrict__ dst,
                         const float* __restrict__ AS, const float* __restrict__ AD,
                         float* __restrict__ EB, float* __restrict__ MX, int nE)
{
    int i = blockIdx.x * blockDim.x + threadIdx.x;
    if (i >= nE * H4) return;
    int e = i >> 2, h = i & 3;
    int s = src[e], d = dst[e];
    float v = AS[(size_t)s * H4 + h] + AD[(size_t)d * H4 + h];
    v = (v > 0.f) ? v : 0.2f * v;
    EB[i] = v;
    atomicMaxF(&MX[(size_t)d * H4 + h], v);
}

__global__ void fix_m(float* __restrict__ MX, int n)
{
    int i = blockIdx.x * blockDim.x + threadIdx.x;
    if (i >= n) return;
    float v = MX[i];
    if (!(v > -1e37f)) MX[i] = 0.f;        // nodes with no in-edges: -inf -> 0
}

// pass 2: ex = exp(e - m[dst]); segment sum over dst; EB overwritten with ex
__global__ void edge_expsum(const int* __restrict__ dst,
                            const float* __restrict__ MX,
                            float* __restrict__ EB, float* __restrict__ DEN, int nE)
{
    int i = blockIdx.x * blockDim.x + threadIdx.x;
    if (i >= nE * H4) return;
    int e = i >> 2, h = i & 3;
    int d = dst[e];
    float ex = __expf(EB[i] - MX[(size_t)d * H4 + h]);
    EB[i] = ex;
    atomicAdd(&DEN[(size_t)d * H4 + h], ex);
}

// pass 3: out[dst, h, c] += xs[src, h, c] * alpha.  One block per edge,
// 256 threads = 256 channels (coalesced loads, L2-resident atomics).
__global__ __launch_bounds__(256) void edge_scatter(
    const int* __restrict__ src, const int* __restrict__ dst,
    const float* __restrict__ XS, const float* __restrict__ EB,
    const float* __restrict__ DEN, float* __restrict__ OUT)
{
    int e = blockIdx.x;
    int t = threadIdx.x;
    int h = t >> 6;
    int s = src[e], d = dst[e];
    float alpha = EB[(size_t)e * H4 + h] /
                  (DEN[(size_t)d * H4 + h] + 1e-16f);
    atomicAdd(&OUT[(size_t)d * D1 + t], XS[(size_t)s * D1 + t] * alpha);
}

// h = relu(LN(conv_out + b + lin_skip))     (one block of 256 per node)
__global__ __launch_bounds__(256) void epilogue_ln(
    const float* __restrict__ OUT, const float* __restrict__ bconv,
    const float* __restrict__ HL, const float* __restrict__ gamma,
    const float* __restrict__ beta, float* __restrict__ HN)
{
    __shared__ float red[256];
    int n = blockIdx.x, t = threadIdx.x;
    float v = OUT[(size_t)n * D1 + t] + bconv[t] + HL[(size_t)n * D1 + t];
    red[t] = v; __syncthreads();
    for (int s2 = 128; s2 > 0; s2 >>= 1) {
        if (t < s2) red[t] += red[t + s2];
        __syncthreads();
    }
    float mu = red[0] * (1.f / 256.f);
    __syncthreads();
    float dv = v - mu;
    red[t] = dv * dv; __syncthreads();
    for (int s2 = 128; s2 > 0; s2 >>= 1) {
        if (t < s2) red[t] += red[t + s2];
        __syncthreads();
    }
    float var = red[0] * (1.f / 256.f);
    float y = dv * rsqrtf(var + 1e-5f) * gamma[t] + beta[t];
    HN[(size_t)n * D1 + t] = fmaxf(y, 0.f);
}

// final: y = conv_out + b + lin_skip
__global__ void epilogue_add(const float* __restrict__ OUT,
                             const float* __restrict__ bconv,
                             const float* __restrict__ HL,
                             float* __restrict__ Y, long total)
{
    long i = (long)blockIdx.x * blockDim.x + threadIdx.x;
    long s = (long)gridDim.x * blockDim.x;
    for (; i < total; i += s)
        Y[i] = OUT[i] + bconv[i & (D1 - 1)] + HL[i];
}

// ---------------------------------------------------------------------------
extern "C" void kernel_launch(void* const* d_in, const int* in_sizes, int n_in,
                              void* d_out, int out_size, void* d_ws, size_t ws_size,
                              hipStream_t stream)
{
    const int DIN = 128;
    const int N = in_sizes[0] / DIN;
    const int E = in_sizes[1] / 2;

    const float* x     = (const float*)d_in[0];
    const int*   ei    = (const int*)  d_in[1];
    const float* Wsrc1 = (const float*)d_in[2];
    const float* Wdst1 = (const float*)d_in[3];
    const float* atts1 = (const float*)d_in[4];
    const float* attd1 = (const float*)d_in[5];
    const float* b1    = (const float*)d_in[6];
    const float* Wlin1 = (const float*)d_in[7];
    const float* blin1 = (const float*)d_in[8];
    const float* gamma = (const float*)d_in[9];
    const float* beta  = (const float*)d_in[10];
    const float* Wsrc2 = (const float*)d_in[11];
    const float* Wdst2 = (const float*)d_in[12];
    const float* atts2 = (const float*)d_in[13];
    const float* attd2 = (const float*)d_in[14];
    const float* b2    = (const float*)d_in[15];
    const float* Wlin2 = (const float*)d_in[16];
    const float* blin2 = (const float*)d_in[17];

    const int* srcI = ei;
    const int* dstI = ei + E;

    // workspace layout (fp32), reused across both layers
    float* ws  = (float*)d_ws;
    size_t NB  = (size_t)N * D1;
    float* XS  = ws;             // N*256
    float* HL  = XS + NB;        // N*256
    float* OUT = HL + NB;        // N*256
    float* HN  = OUT + NB;       // N*256
    float* EB  = HN + NB;        // E*4
    float* AS  = EB + (size_t)E * H4;
    float* AD  = AS + (size_t)N * H4;
    float* MX  = AD + (size_t)N * H4;
    float* DEN = MX + (size_t)N * H4;
    float* VD  = DEN + (size_t)N * H4;   // 256*4

    const int Mtiles = N / 16;                       // N = 50000 -> 3125
    dim3 ggrid((Mtiles + 7) / 8, D1 / 16);
    const int ehThreads = E * H4;
    const int ehBlocks  = (ehThreads + 255) / 256;
    const int nhBlocks  = (N * H4 + 255) / 256;
    const int waveBlocks = (N + 7) / 8;              // 8 waves/block, 1 wave/node

    const float NEG_INF = -__builtin_huge_valf();

    // ============================ layer 1 ============================
    wmma_gemm_f32<<<ggrid, 256, 0, stream>>>(x, Wsrc1, nullptr, XS, Mtiles, DIN);
    wmma_gemm_f32<<<ggrid, 256, 0, stream>>>(x, Wlin1, blin1,   HL, Mtiles, DIN);
    collapse_att<<<(DIN * H4 + 255) / 256, 256, 0, stream>>>(Wdst1, attd1, VD, DIN);
    gemv_ad<<<waveBlocks, 256, 0, stream>>>(x, VD, AD, N, DIN);
    attn_src_scores<<<waveBlocks, 256, 0, stream>>>(XS, atts1, AS, N);

    fill_f32<<<2048, 256, 0, stream>>>(MX, NEG_INF, (long)N * H4);
    fill_f32<<<2048, 256, 0, stream>>>(DEN, 0.f, (long)N * H4);
    fill_f32<<<2048, 256, 0, stream>>>(OUT, 0.f, (long)NB);

    edge_max<<<ehBlocks, 256, 0, stream>>>(srcI, dstI, AS, AD, EB, MX, E);
    fix_m<<<nhBlocks, 256, 0, stream>>>(MX, N * H4);
    edge_expsum<<<ehBlocks, 256, 0, stream>>>(dstI, MX, EB, DEN, E);
    edge_scatter<<<E, 256, 0, stream>>>(srcI, dstI, XS, EB, DEN, OUT);

    epilogue_ln<<<N, 256, 0, stream>>>(OUT, b1, HL, gamma, beta, HN);

    // ============================ layer 2 ============================
    wmma_gemm_f32<<<ggrid, 256, 0, stream>>>(HN, Wsrc2, nullptr, XS, Mtiles, D1);
    wmma_gemm_f32<<<ggrid, 256, 0, stream>>>(HN, Wlin2, blin2,   HL, Mtiles, D1);
    collapse_att<<<(D1 * H4 + 255) / 256, 256, 0, stream>>>(Wdst2, attd2, VD, D1);
    gemv_ad<<<waveBlocks, 256, 0, stream>>>(HN, VD, AD, N, D1);
    attn_src_scores<<<waveBlocks, 256, 0, stream>>>(XS, atts2, AS, N);

    fill_f32<<<2048, 256, 0, stream>>>(MX, NEG_INF, (long)N * H4);
    fill_f32<<<2048, 256, 0, stream>>>(DEN, 0.f, (long)N * H4);
    fill_f32<<<2048, 256, 0, stream>>>(OUT, 0.f, (long)NB);

    edge_max<<<ehBlocks, 256, 0, stream>>>(srcI, dstI, AS, AD, EB, MX, E);
    fix_m<<<nhBlocks, 256, 0, stream>>>(MX, N * H4);
    edge_expsum<<<ehBlocks, 256, 0, stream>>>(dstI, MX, EB, DEN, E);
    edge_scatter<<<E, 256, 0, stream>>>(srcI, dstI, XS, EB, DEN, OUT);

    epilogue_add<<<2048, 256, 0, stream>>>(OUT, b2, HL, (float*)d_out, (long)NB);
}